// IdentityUnitaryR3Ansatz_batch_18846316495454
// MI455X (gfx1250) — compile-verified
//
#include <hip/hip_runtime.h>

typedef __attribute__((ext_vector_type(2))) float v2f;
typedef __attribute__((ext_vector_type(8))) float v8f;

#define NQ      7
#define DIM     128
#define NLAYERS 8

__device__ __forceinline__ float2 cmul(float2 a, float2 b) {
    return make_float2(a.x * b.x - a.y * b.y, a.x * b.y + a.y * b.x);
}

__device__ __forceinline__ v8f wmma_f32(v2f a, v2f b, v8f c) {
    // D = A(16x4) * B(4x16) + C, f32 everywhere.
    return __builtin_amdgcn_wmma_f32_16x16x4_f32(false, a, false, b, (short)0, c, false, false);
}

__global__ __launch_bounds__(32) void qsim_r3_wmma_kernel(
    const float* __restrict__ state_in,   // (TB, 128) f32
    const float* __restrict__ params,     // (TB, 8, 7, 3) f32
    float* __restrict__ out)              // (TB, 128) complex64 interleaved
{
    __shared__ float gre[NQ][4], gim[NQ][4];   // per-layer 2x2 gates, entry idx = row*2+col
    __shared__ float sre[2][DIM], sim[2][DIM]; // double-buffered state (8x16 view)
    __shared__ float tre[DIM], tim[DIM];       // intermediate T = S @ B^T (rows 0..7)

    const int bid  = blockIdx.x;
    const int lane = threadIdx.x;
    const int l    = lane & 15;
    const int hi   = lane >> 4;

    // Load real input state, imag = 0
    #pragma unroll
    for (int k = 0; k < 4; ++k) {
        const int idx = lane * 4 + k;
        sre[0][idx] = state_in[bid * DIM + idx];
        sim[0][idx] = 0.0f;
    }
    __syncthreads();

    for (int layer = 0; layer < NLAYERS; ++layer) {
        const int cur = layer & 1;
        const int nxt = cur ^ 1;

        // --- 7 lanes compute the per-qubit R3 gates for this layer ---
        if (lane < NQ) {
            const float* p = params + ((size_t)(bid * NLAYERS + layer) * NQ + lane) * 3;
            const float omega = p[0], theta = p[1], phi = p[2];
            const float hs = 0.5f * (phi + omega);   // e1 = exp(-i*hs)
            const float hd = 0.5f * (phi - omega);   // e2 = exp(+i*hd)
            const float ht = 0.5f * theta;
            const float c1 = __cosf(hs), s1 = __sinf(hs);
            const float c2 = __cosf(hd), s2 = __sinf(hd);
            const float c  = __cosf(ht), s  = __sinf(ht);
            gre[lane][0] =  c1 * c;  gim[lane][0] = -s1 * c;   // e1*c
            gre[lane][1] = -c2 * s;  gim[lane][1] = -s2 * s;   // -e2*s
            gre[lane][2] =  c2 * s;  gim[lane][2] = -s2 * s;   // conj(e2)*s
            gre[lane][3] =  c1 * c;  gim[lane][3] =  s1 * c;   // conj(e1)*c
        }
        __syncthreads();

        // --- Per-lane gate rows needed for kron elements ---
        float2 g0c[2], g1c[2], g2c[2], g3c[2], g4c[2], g5c[2], g6c[2];
        {
            const int la = l & 7;
            const int r0 = (la >> 2) & 1, r1 = (la >> 1) & 1, r2 = la & 1;
            const int r3 = (l >> 3) & 1, r4 = (l >> 2) & 1, r5 = (l >> 1) & 1, r6 = l & 1;
            #pragma unroll
            for (int j = 0; j < 2; ++j) {
                g0c[j] = make_float2(gre[0][r0 * 2 + j], gim[0][r0 * 2 + j]);
                g1c[j] = make_float2(gre[1][r1 * 2 + j], gim[1][r1 * 2 + j]);
                g2c[j] = make_float2(gre[2][r2 * 2 + j], gim[2][r2 * 2 + j]);
                g3c[j] = make_float2(gre[3][r3 * 2 + j], gim[3][r3 * 2 + j]);
                g4c[j] = make_float2(gre[4][r4 * 2 + j], gim[4][r4 * 2 + j]);
                g5c[j] = make_float2(gre[5][r5 * 2 + j], gim[5][r5 * 2 + j]);
                g6c[j] = make_float2(gre[6][r6 * 2 + j], gim[6][r6 * 2 + j]);
            }
        }

        const float rowMask = (l < 8) ? 1.0f : 0.0f;  // pad rows 8..15 of S and A with zeros
        const int   lr      = (l & 7) * 16;

        // ---- T = S @ B^T   (complex 16x16x16, chained K chunks of 4) ----
        v8f accRe = {0.f,0.f,0.f,0.f,0.f,0.f,0.f,0.f};
        v8f accIm = {0.f,0.f,0.f,0.f,0.f,0.f,0.f,0.f};
        #pragma unroll
        for (int k0 = 0; k0 < 4; ++k0) {
            const int m0 = 4 * k0 + 2 * hi;
            // A-operand: rows of S (state), row l, cols m0/m0+1
            v2f aSr, aSi;
            aSr.x = rowMask * sre[cur][lr + m0];
            aSr.y = rowMask * sre[cur][lr + m0 + 1];
            aSi.x = rowMask * sim[cur][lr + m0];
            aSi.y = rowMask * sim[cur][lr + m0 + 1];
            const v2f aNSi = -aSi;
            // B-operand: (B^T)[m][l] = Bmat[l][m] = kron(g3..g6)[l][m]
            const float2 pre = cmul(cmul(g3c[(m0 >> 3) & 1], g4c[(m0 >> 2) & 1]),
                                    g5c[(m0 >> 1) & 1]);
            const float2 b0 = cmul(pre, g6c[0]);   // m0 (even -> g6 col 0)
            const float2 b1 = cmul(pre, g6c[1]);   // m0+1
            v2f bRe, bIm;
            bRe.x = b0.x; bRe.y = b1.x;
            bIm.x = b0.y; bIm.y = b1.y;
            accRe = wmma_f32(aSr,  bRe, accRe);   //  Sr*BrT
            accRe = wmma_f32(aNSi, bIm, accRe);   // -Si*BiT
            accIm = wmma_f32(aSr,  bIm, accIm);   //  Sr*BiT
            accIm = wmma_f32(aSi,  bRe, accIm);   //  Si*BrT
        }

        // C/D layout -> LDS (only rows 0..7 are nonzero, owned by hi==0 lanes)
        if (hi == 0) {
            #pragma unroll
            for (int v = 0; v < 8; ++v) {
                tre[v * 16 + l] = accRe[v];
                tim[v * 16 + l] = accIm[v];
            }
        }
        __syncthreads();

        // ---- S' = A_pad @ T   (complex, effective K = 8) ----
        v8f ar2 = {0.f,0.f,0.f,0.f,0.f,0.f,0.f,0.f};
        v8f ai2 = {0.f,0.f,0.f,0.f,0.f,0.f,0.f,0.f};
        #pragma unroll
        for (int k0 = 0; k0 < 2; ++k0) {
            const int m0 = 4 * k0 + 2 * hi;
            // A-operand: Amat[l][m] = kron(g0,g1,g2)[l][m], rows >=8 zero
            const float2 preA = cmul(g0c[(m0 >> 2) & 1], g1c[(m0 >> 1) & 1]);
            const float2 a0 = cmul(preA, g2c[0]);
            const float2 a1 = cmul(preA, g2c[1]);
            v2f aRe, aIm;
            aRe.x = rowMask * a0.x; aRe.y = rowMask * a1.x;
            aIm.x = rowMask * a0.y; aIm.y = rowMask * a1.y;
            const v2f aNIm = -aIm;
            // B-operand: T rows m0/m0+1, col l
            v2f bTr, bTi;
            bTr.x = tre[m0 * 16 + l];       bTr.y = tre[(m0 + 1) * 16 + l];
            bTi.x = tim[m0 * 16 + l];       bTi.y = tim[(m0 + 1) * 16 + l];
            ar2 = wmma_f32(aRe,  bTr, ar2);
            ar2 = wmma_f32(aNIm, bTi, ar2);
            ai2 = wmma_f32(aRe,  bTi, ai2);
            ai2 = wmma_f32(aIm,  bTr, ai2);
        }
        __syncthreads();

        // ---- Entangler permutation: scatter into next buffer ----
        // even layer (E):    dest = i ^ (i>>1)
        // odd layer (E^T):   dest = prefixXor(i)
        if (hi == 0) {
            #pragma unroll
            for (int v = 0; v < 8; ++v) {
                const int i = v * 16 + l;
                int dest;
                if (layer & 1) {
                    int x = i; x ^= x >> 1; x ^= x >> 2; x ^= x >> 4;
                    dest = x & 127;
                } else {
                    dest = (i ^ (i >> 1)) & 127;
                }
                sre[nxt][dest] = ar2[v];
                sim[nxt][dest] = ai2[v];
            }
        }
        __syncthreads();
    }

    // Final state lives in buffer 0 after 8 layers; write complex64 interleaved.
    float2* o = (float2*)out;
    #pragma unroll
    for (int k = 0; k < 4; ++k) {
        const int j = lane * 4 + k;
        o[bid * DIM + j] = make_float2(sre[0][j], sim[0][j]);
    }
}

extern "C" void kernel_launch(void* const* d_in, const int* in_sizes, int n_in,
                              void* d_out, int out_size, void* d_ws, size_t ws_size,
                              hipStream_t stream) {
    const float* state  = (const float*)d_in[0];   // input_state (TB, 128) f32
    const float* params = (const float*)d_in[1];   // params (T,B,8,7,3) f32
    float* out = (float*)d_out;                    // complex64 (TB, 128) interleaved
    const int nbatch = in_sizes[0] / DIM;          // TB = 1024
    qsim_r3_wmma_kernel<<<nbatch, 32, 0, stream>>>(state, params, out);
}